// BinaryCrossEntropyP_67843303408220
// MI455X (gfx1250) — compile-verified
//
#include <hip/hip_runtime.h>
#include <stdint.h>

// Problem constants (fixed by the reference).
#define B_ROWS 8192
#define L_COLS 4096
static constexpr long long TOTAL   = (long long)B_ROWS * L_COLS;   // 33.5M elements
static constexpr int       NSEG    = (int)(TOTAL / 32);            // 1,048,576 mask words
static constexpr int       RBLOCK  = 256;
static constexpr int       RGRID   = NSEG / RBLOCK;                // 4096 blocks

typedef float v2f __attribute__((ext_vector_type(2)));
typedef float v8f __attribute__((ext_vector_type(8)));

__device__ __forceinline__ uint32_t hash32(uint32_t x) {
    x ^= x >> 16; x *= 0x7feb352dU;
    x ^= x >> 15; x *= 0x846ca68bU;
    x ^= x >> 16;
    return x;
}

// ---------------- pass 0: zero a u32 region ----------------
__global__ void zero_u32(uint32_t* __restrict__ p, int n) {
    int i = blockIdx.x * blockDim.x + threadIdx.x;
    if (i < n) p[i] = 0u;
}

// ---------------- pass 1: per-column positive counts ----------------
// grid = (L/256, B/512), block = 256. Lanes walk consecutive columns -> coalesced.
__global__ void count_pos(const float* __restrict__ target, uint32_t* __restrict__ nbpos) {
    int col = blockIdx.x * 256 + threadIdx.x;
    int r0  = blockIdx.y * 512;
    uint32_t c = 0;
    for (int r = 0; r < 512; ++r)
        c += (target[(size_t)(r0 + r) * L_COLS + col] > 0.5f) ? 1u : 0u;
    if (c) atomicAdd(&nbpos[col], c);
}

// ---------------- pass 2: negative sampling into a bitmask ----------------
// One thread per column; rejection-sample rows (with replacement) uniformly over
// zero entries (~95% acceptance), scatter bits with idempotent atomicOr.
__global__ void sample_neg(const float* __restrict__ target,
                           const uint32_t* __restrict__ nbpos,
                           uint32_t* __restrict__ mask) {
    int col = blockIdx.x * 256 + threadIdx.x;
    uint32_t n = nbpos[col];                      // NEG_RATE == 1.0
    for (uint32_t s = 0; s < n; ++s) {
        uint32_t st = (uint32_t)col * 0x9E3779B9u ^ (s * 0x85EBCA6Bu) ^ 0x2C1B3C6DU;
        for (int att = 0; att < 32; ++att) {
            uint32_t r = hash32(st + (uint32_t)att * 0xC2B2AE35u) & (B_ROWS - 1);
            size_t idx = (size_t)r * L_COLS + col;
            if (target[idx] <= 0.5f) {            // zero entry -> accept
                atomicOr(&mask[idx >> 5], 1u << (idx & 31));
                break;
            }
        }
    }
}

// ---------------- pass 3: fused BCE * mask + reduction ----------------
// Each thread owns one 32-element segment (one mask word), streamed with
// float4 loads. Wave partials are collapsed with V_WMMA_F32_16X16X4_F32
// (B = all-ones): D[m][n] = acc[m] + acc[m+16], lanes 0/16 carry the two
// half-sums of the wave.
__global__ void bce_reduce(const float* __restrict__ P, const float* __restrict__ T,
                           const uint32_t* __restrict__ mask,
                           float* __restrict__ blockSums) {
    int seg = blockIdx.x * blockDim.x + threadIdx.x;
    size_t base = (size_t)seg * 32;
    uint32_t mw = mask[seg];

    const float4* p4 = (const float4*)(P + base);
    const float4* t4 = (const float4*)(T + base);

    float acc = 0.0f;
#pragma unroll
    for (int c = 0; c < 8; ++c) {
        float4 pv = p4[c];
        float4 tv = t4[c];
        float pe[4] = {pv.x, pv.y, pv.z, pv.w};
        float te[4] = {tv.x, tv.y, tv.z, tv.w};
#pragma unroll
        for (int j = 0; j < 4; ++j) {
            int e = c * 4 + j;
            float tt = te[j];
            float m = fmaxf(tt, (float)((mw >> e) & 1u));  // mask = pos OR sampled-neg
            if (m != 0.0f) {                                // ~90% of lanes skip the logs
                float pp  = pe[j];
                float lp  = fmaxf(__logf(pp),        -100.0f);
                float l1p = fmaxf(__logf(1.0f - pp), -100.0f);
                acc -= tt * lp + (1.0f - tt) * l1p;         // m==1 here
            }
        }
    }

    // Wave32 reduction on the matrix unit (fp32, single instruction).
    v2f a; a.x = acc;  a.y = 0.0f;      // A[m][0]=lane m, A[m][2]=lane m+16, K=1,3 zero
    v2f b; b.x = 1.0f; b.y = 1.0f;      // B = all ones (layout-agnostic)
    v8f cz = {};
    v8f d = __builtin_amdgcn_wmma_f32_16x16x4_f32(
        /*neg_a=*/false, a, /*neg_b=*/false, b,
        /*c_mod=*/(short)0, cz, /*reuse_a=*/false, /*reuse_b=*/false);
    float s = d[0] + d[1] + d[2] + d[3] + d[4] + d[5] + d[6] + d[7];

    __shared__ float lds[16];           // 8 waves * 2 half-sums
    int lane = threadIdx.x & 31;
    int wid  = threadIdx.x >> 5;
    if ((lane & 15) == 0) lds[wid * 2 + (lane >> 4)] = s;
    __syncthreads();
    if (threadIdx.x == 0) {
        float tot = 0.0f;
#pragma unroll
        for (int i = 0; i < 16; ++i) tot += lds[i];
        blockSums[blockIdx.x] = tot;
    }
}

// ---------------- pass 4: deterministic final sum + mean ----------------
__global__ void finalize(const float* __restrict__ blockSums, float* __restrict__ out) {
    __shared__ float lds[256];
    float s = 0.0f;
    for (int i = threadIdx.x; i < RGRID; i += 256) s += blockSums[i];
    lds[threadIdx.x] = s;
    __syncthreads();
    if (threadIdx.x == 0) {
        float tot = 0.0f;
        for (int i = 0; i < 256; ++i) tot += lds[i];
        out[0] = tot / (float)TOTAL;
    }
}

extern "C" void kernel_launch(void* const* d_in, const int* in_sizes, int n_in,
                              void* d_out, int out_size, void* d_ws, size_t ws_size,
                              hipStream_t stream) {
    (void)in_sizes; (void)n_in; (void)out_size; (void)ws_size;
    const float* inputs = (const float*)d_in[0];   // predicted probs [B, L]
    const float* target = (const float*)d_in[1];   // binary labels  [B, L]
    float* out = (float*)d_out;

    // Workspace layout (bytes):
    //   [0,     16384) : nb_pos   (L_COLS  u32)
    //   [16384, 32768) : blockSums(RGRID   f32)
    //   [32768, +4MB ) : mask bits(NSEG    u32)
    uint8_t*  ws        = (uint8_t*)d_ws;
    uint32_t* nbpos     = (uint32_t*)(ws);
    float*    blockSums = (float*)(ws + 16384);
    uint32_t* mask      = (uint32_t*)(ws + 32768);

    // zero counts + mask each launch (graph-replay safe, deterministic)
    zero_u32<<<(L_COLS + 255) / 256, 256, 0, stream>>>(nbpos, L_COLS);
    zero_u32<<<(NSEG + 255) / 256, 256, 0, stream>>>(mask, NSEG);

    dim3 cgrid(L_COLS / 256, B_ROWS / 512);
    count_pos<<<cgrid, 256, 0, stream>>>(target, nbpos);

    sample_neg<<<L_COLS / 256, 256, 0, stream>>>(target, nbpos, mask);

    bce_reduce<<<RGRID, RBLOCK, 0, stream>>>(inputs, target, mask, blockSums);

    finalize<<<1, 256, 0, stream>>>(blockSums, out);
}